// LaCTMLPSWIGLULayer_35029753266577
// MI455X (gfx1250) — compile-verified
//
#include <hip/hip_runtime.h>

// LaCT MLP-SwiGLU layer forward for MI455X (gfx1250), pure f32 using
// V_WMMA_F32_16X16X4_F32 for every GEMM-shaped contraction.
// Fragment loads are half-wave pre-offset b64 loads (lane needs exactly
// A[m][k + 2*hi .. k+2*hi+1]) and the K-loops are double-buffered so VMEM
// overlaps the WMMA pipe.

typedef float v2f __attribute__((ext_vector_type(2)));
typedef float v8f __attribute__((ext_vector_type(8)));

#define B_    2
#define S_    2048
#define H_    2048
#define NAH_  16
#define AHD_  128
#define NFH_  4
#define FHD_  512
#define WIN_  1024
#define CHUNK_ 512
#define BS_   (B_*S_)
#define EPS_  1e-6f
#define BASE_LR_INV_ (-6.9072552f)      /* 0.001 + log(-expm1(-0.001)) */
#define LOG_ROPE_BASE_ 13.122363377404329f /* ln(500000) */
#define ATTN_SCALE_ 0.088388347648318447f  /* 1/sqrt(128) */

__device__ __forceinline__ v8f vzero8() {
  v8f z = {0.f,0.f,0.f,0.f,0.f,0.f,0.f,0.f};
  return z;
}

__device__ __forceinline__ v2f ld2(const float* p) {
  return *(const v2f*)p;   // 8-byte aligned global_load_b64
}

__device__ __forceinline__ v8f wmma_f32(v2f a, v2f b, v8f c) {
  // D = A(16x4) * B(4x16) + C(16x16), full f32
  return __builtin_amdgcn_wmma_f32_16x16x4_f32(false, a, false, b, (short)0, c,
                                               false, false);
}

__device__ __forceinline__ float silu_f(float x) {
  return x / (1.0f + __expf(-x));
}

__device__ __forceinline__ float half_max16(float v) {
  v = fmaxf(v, __shfl_xor(v, 1, 32));
  v = fmaxf(v, __shfl_xor(v, 2, 32));
  v = fmaxf(v, __shfl_xor(v, 4, 32));
  v = fmaxf(v, __shfl_xor(v, 8, 32));
  return v;
}
__device__ __forceinline__ float half_sum16(float v) {
  v += __shfl_xor(v, 1, 32);
  v += __shfl_xor(v, 2, 32);
  v += __shfl_xor(v, 4, 32);
  v += __shfl_xor(v, 8, 32);
  return v;
}

// ---------------------------------------------------------------------------
// GEMM: C = A[M,K] @ W[N,K]^T   (x @ W^T pattern; all projections + TTT apply)
// Batched over z = b*4+h via base strides. Epilogue: 0=store, 1=silu, 2=add R.
// Wave computes a 16x64 strip; block (32,4) covers 64x64. Double-buffered K.
// ---------------------------------------------------------------------------
__global__ __launch_bounds__(128)
void gemm_xwt(const float* __restrict__ A, long lda, long aBS, long aHS,
              const float* __restrict__ W, long ldw, long wBS, long wHS,
              float* __restrict__ C, long ldc, long cBS, long cHS,
              const float* __restrict__ R, long ldr, int K, int ep)
{
  const int lane = threadIdx.x, wid = threadIdx.y;
  const int ln = lane & 15, hi = lane >> 4;
  const int zb = blockIdx.z >> 2, zh = blockIdx.z & 3;
  const float* Ab = A + (size_t)zb * aBS + (size_t)zh * aHS;
  const float* Wb = W + (size_t)zb * wBS + (size_t)zh * wHS;
  float*       Cb = C + (size_t)zb * cBS + (size_t)zh * cHS;
  const int m0 = blockIdx.x * 64 + wid * 16;
  const int n0 = blockIdx.y * 64;

  v8f acc[4];
  #pragma unroll
  for (int nt = 0; nt < 4; ++nt) acc[nt] = vzero8();

  const float* arow = Ab + (size_t)(m0 + ln) * lda + 2 * hi;
  const float* wr[4];
  #pragma unroll
  for (int nt = 0; nt < 4; ++nt)
    wr[nt] = Wb + (size_t)(n0 + nt * 16 + ln) * ldw + 2 * hi;

  v2f a_c = ld2(arow);
  v2f w_c[4];
  #pragma unroll
  for (int nt = 0; nt < 4; ++nt) w_c[nt] = ld2(wr[nt]);

  for (int k = 0; k < K - 4; k += 4) {
    v2f a_n = ld2(arow + k + 4);
    v2f w_n[4];
    #pragma unroll
    for (int nt = 0; nt < 4; ++nt) w_n[nt] = ld2(wr[nt] + k + 4);
    #pragma unroll
    for (int nt = 0; nt < 4; ++nt) acc[nt] = wmma_f32(a_c, w_c[nt], acc[nt]);
    a_c = a_n;
    #pragma unroll
    for (int nt = 0; nt < 4; ++nt) w_c[nt] = w_n[nt];
  }
  #pragma unroll
  for (int nt = 0; nt < 4; ++nt) acc[nt] = wmma_f32(a_c, w_c[nt], acc[nt]);

  #pragma unroll
  for (int nt = 0; nt < 4; ++nt) {
    #pragma unroll
    for (int r = 0; r < 8; ++r) {
      const int m = m0 + r + hi * 8;
      const int n = n0 + nt * 16 + ln;
      float v = acc[nt][r];
      if (ep == 1)      v = silu_f(v);
      else if (ep == 2) v += R[(size_t)m * ldr + n];
      Cb[(size_t)m * ldc + n] = v;
    }
  }
}

// ---------------------------------------------------------------------------
// GEMM: C = A[M,K] @ B[K,N]  (dhid = v @ W1), double-buffered K.
// ---------------------------------------------------------------------------
__global__ __launch_bounds__(128)
void gemm_xw(const float* __restrict__ A, long lda, long aBS, long aHS,
             const float* __restrict__ Bm, long ldb, long bBS, long bHS,
             float* __restrict__ C, long ldc, long cBS, long cHS, int K)
{
  const int lane = threadIdx.x, wid = threadIdx.y;
  const int ln = lane & 15, hi = lane >> 4;
  const int zb = blockIdx.z >> 2, zh = blockIdx.z & 3;
  const float* Ab = A + (size_t)zb * aBS + (size_t)zh * aHS;
  const float* Bb = Bm + (size_t)zb * bBS + (size_t)zh * bHS;
  float*       Cb = C + (size_t)zb * cBS + (size_t)zh * cHS;
  const int m0 = blockIdx.x * 64 + wid * 16;
  const int n0 = blockIdx.y * 64;

  v8f acc[4];
  #pragma unroll
  for (int nt = 0; nt < 4; ++nt) acc[nt] = vzero8();

  const float* arow = Ab + (size_t)(m0 + ln) * lda + 2 * hi;

  v2f a_c = ld2(arow);
  v2f b_c[4];
  {
    const float* br0 = Bb + (size_t)(2 * hi) * ldb + n0 + ln;
    const float* br1 = br0 + ldb;
    #pragma unroll
    for (int nt = 0; nt < 4; ++nt) { b_c[nt].x = br0[nt*16]; b_c[nt].y = br1[nt*16]; }
  }
  for (int k = 0; k < K - 4; k += 4) {
    v2f a_n = ld2(arow + k + 4);
    v2f b_n[4];
    const float* br0 = Bb + (size_t)(k + 4 + 2 * hi) * ldb + n0 + ln;
    const float* br1 = br0 + ldb;
    #pragma unroll
    for (int nt = 0; nt < 4; ++nt) { b_n[nt].x = br0[nt*16]; b_n[nt].y = br1[nt*16]; }
    #pragma unroll
    for (int nt = 0; nt < 4; ++nt) acc[nt] = wmma_f32(a_c, b_c[nt], acc[nt]);
    a_c = a_n;
    #pragma unroll
    for (int nt = 0; nt < 4; ++nt) b_c[nt] = b_n[nt];
  }
  #pragma unroll
  for (int nt = 0; nt < 4; ++nt) acc[nt] = wmma_f32(a_c, b_c[nt], acc[nt]);

  #pragma unroll
  for (int nt = 0; nt < 4; ++nt)
    #pragma unroll
    for (int r = 0; r < 8; ++r)
      Cb[(size_t)(m0 + r + hi * 8) * ldc + n0 + nt * 16 + ln] = acc[nt][r];
}

// ---------------------------------------------------------------------------
// GEMM: C[M,N] += A[K,M]^T @ B[K,N], with optional per-k (per-token) row
// scales on A or B (the lr1/lr0/lr2 learning-rate factors). Double-buffered.
// ---------------------------------------------------------------------------
__global__ __launch_bounds__(128)
void gemm_atb(const float* __restrict__ A, long lda, long aBS, long aHS,
              const float* __restrict__ Bm, long ldb, long bBS, long bHS,
              float* __restrict__ C, long ldc, long cBS, long cHS,
              const float* sA, const float* sB, long sBS, long sHS, long ss,
              int K)
{
  const int lane = threadIdx.x, wid = threadIdx.y;
  const int ln = lane & 15, hi = lane >> 4;
  const int zb = blockIdx.z >> 2, zh = blockIdx.z & 3;
  const float* Ab = A + (size_t)zb * aBS + (size_t)zh * aHS;
  const float* Bb = Bm + (size_t)zb * bBS + (size_t)zh * bHS;
  float*       Cb = C + (size_t)zb * cBS + (size_t)zh * cHS;
  const float* sAb = sA ? (sA + (size_t)zb * sBS + (size_t)zh * sHS) : nullptr;
  const float* sBb = sB ? (sB + (size_t)zb * sBS + (size_t)zh * sHS) : nullptr;
  const int m0 = blockIdx.x * 64 + wid * 16;
  const int n0 = blockIdx.y * 64;

  v8f acc[4];
  #pragma unroll
  for (int nt = 0; nt < 4; ++nt) acc[nt] = vzero8();

  v2f a_c, b_c[4];
  {
    const int k0 = 2 * hi;
    float a0 = Ab[(size_t)k0 * lda + m0 + ln];
    float a1 = Ab[(size_t)(k0 + 1) * lda + m0 + ln];
    if (sAb) { a0 *= sAb[(size_t)k0 * ss]; a1 *= sAb[(size_t)(k0 + 1) * ss]; }
    a_c.x = a0; a_c.y = a1;
    const float s0 = sBb ? sBb[(size_t)k0 * ss] : 1.f;
    const float s1 = sBb ? sBb[(size_t)(k0 + 1) * ss] : 1.f;
    const float* br0 = Bb + (size_t)k0 * ldb + n0 + ln;
    const float* br1 = br0 + ldb;
    #pragma unroll
    for (int nt = 0; nt < 4; ++nt) { b_c[nt].x = br0[nt*16]*s0; b_c[nt].y = br1[nt*16]*s1; }
  }
  for (int k = 0; k < K - 4; k += 4) {
    const int k0 = k + 4 + 2 * hi;
    float a0 = Ab[(size_t)k0 * lda + m0 + ln];
    float a1 = Ab[(size_t)(k0 + 1) * lda + m0 + ln];
    if (sAb) { a0 *= sAb[(size_t)k0 * ss]; a1 *= sAb[(size_t)(k0 + 1) * ss]; }
    v2f a_n; a_n.x = a0; a_n.y = a1;
    const float s0 = sBb ? sBb[(size_t)k0 * ss] : 1.f;
    const float s1 = sBb ? sBb[(size_t)(k0 + 1) * ss] : 1.f;
    const float* br0 = Bb + (size_t)k0 * ldb + n0 + ln;
    const float* br1 = br0 + ldb;
    v2f b_n[4];
    #pragma unroll
    for (int nt = 0; nt < 4; ++nt) { b_n[nt].x = br0[nt*16]*s0; b_n[nt].y = br1[nt*16]*s1; }
    #pragma unroll
    for (int nt = 0; nt < 4; ++nt) acc[nt] = wmma_f32(a_c, b_c[nt], acc[nt]);
    a_c = a_n;
    #pragma unroll
    for (int nt = 0; nt < 4; ++nt) b_c[nt] = b_n[nt];
  }
  #pragma unroll
  for (int nt = 0; nt < 4; ++nt) acc[nt] = wmma_f32(a_c, b_c[nt], acc[nt]);

  #pragma unroll
  for (int nt = 0; nt < 4; ++nt)
    #pragma unroll
    for (int r = 0; r < 8; ++r) {
      const size_t idx = (size_t)(m0 + r + hi * 8) * ldc + n0 + nt * 16 + ln;
      Cb[idx] += acc[nt][r];
    }
}

// ---------------------------------------------------------------------------
// RMSNorm(full H) on q/k halves of qkv + half-split RoPE + scatter to
// [B*NAH][S][AHD] layout (also copies v).
// ---------------------------------------------------------------------------
__global__ __launch_bounds__(256)
void rope_split(const float* __restrict__ qkv,
                const float* __restrict__ qw, const float* __restrict__ kw,
                float* __restrict__ qr, float* __restrict__ kr,
                float* __restrict__ vr)
{
  __shared__ float red[256];
  const int t = blockIdx.x;
  const int b = t >> 11, s = t & 2047;
  const int tid = threadIdx.x;
  const float* row = qkv + (size_t)t * 6144;

  float sq = 0.f, sk = 0.f;
  for (int c = tid; c < 2048; c += 256) {
    float v = row[c];        sq += v * v;
    float u = row[2048 + c]; sk += u * u;
  }
  red[tid] = sq; __syncthreads();
  for (int st = 128; st > 0; st >>= 1) { if (tid < st) red[tid] += red[tid + st]; __syncthreads(); }
  const float rq = rsqrtf(red[0] / 2048.f + EPS_);
  __syncthreads();
  red[tid] = sk; __syncthreads();
  for (int st = 128; st > 0; st >>= 1) { if (tid < st) red[tid] += red[tid + st]; __syncthreads(); }
  const float rk = rsqrtf(red[0] / 2048.f + EPS_);

  for (int idx = tid; idx < 1024; idx += 256) {
    const int h = idx >> 6, i = idx & 63;
    const float inv = __expf(-((float)(2 * i) / 128.f) * LOG_ROPE_BASE_);
    const float ang = (float)s * inv;
    float sn, cs; __sincosf(ang, &sn, &cs);
    const int qb = h * 128 + i;
    const size_t ob = ((size_t)(b * NAH_ + h) * S_ + s) * AHD_;
    float x1 = row[qb] * rq * qw[qb];
    float x2 = row[qb + 64] * rq * qw[qb + 64];
    qr[ob + i]      = x1 * cs - x2 * sn;
    qr[ob + 64 + i] = x2 * cs + x1 * sn;
    float y1 = row[2048 + qb] * rk * kw[qb];
    float y2 = row[2048 + qb + 64] * rk * kw[qb + 64];
    kr[ob + i]      = y1 * cs - y2 * sn;
    kr[ob + 64 + i] = y2 * cs + y1 * sn;
  }
  for (int idx = tid; idx < 2048; idx += 256) {
    const int h = idx >> 7, d = idx & 127;
    vr[((size_t)(b * NAH_ + h) * S_ + s) * AHD_ + d] = row[4096 + idx];
  }
}

// ---------------------------------------------------------------------------
// Sliding-window attention, flash style, all-WMMA. One wave per 16-query tile.
// ---------------------------------------------------------------------------
__global__ __launch_bounds__(32)
void attn_swa(const float* __restrict__ Q, const float* __restrict__ Km,
              const float* __restrict__ V, float* __restrict__ O)
{
  __shared__ float pl[16][16];
  const int lane = threadIdx.x;
  const int ln = lane & 15, hi = lane >> 4;
  const int it = blockIdx.x, h = blockIdx.y, b = blockIdx.z;
  const int i0 = it * 16;
  const size_t hoff = (size_t)(b * NAH_ + h) * S_ * AHD_;
  const float* Qh = Q + hoff;
  const float* Kh = Km + hoff;
  const float* Vh = V + hoff;

  v8f acc[8];
  #pragma unroll
  for (int nt = 0; nt < 8; ++nt) acc[nt] = vzero8();
  float mrow[8], lrow[8];
  #pragma unroll
  for (int r = 0; r < 8; ++r) { mrow[r] = -1e30f; lrow[r] = 0.f; }

  const float* qrow = Qh + (size_t)(i0 + ln) * AHD_ + 2 * hi;

  const int jt_lo = (it > 63) ? (it - 63) : 0;
  for (int jt = jt_lo; jt <= it; ++jt) {
    const int j0 = jt * 16;
    v8f s = vzero8();
    const float* krow = Kh + (size_t)(j0 + ln) * AHD_ + 2 * hi;
    #pragma unroll
    for (int k4 = 0; k4 < 32; ++k4) {
      v2f af = ld2(qrow + 4 * k4);
      v2f bf = ld2(krow + 4 * k4);
      s = wmma_f32(af, bf, s);
    }
    float p[8];
    #pragma unroll
    for (int r = 0; r < 8; ++r) {
      const int i = i0 + r + hi * 8;
      const int j = j0 + ln;
      const bool valid = (j <= i) && ((i - j) < WIN_);
      const float sv = valid ? s[r] * ATTN_SCALE_ : -1e30f;
      const float mx = half_max16(sv);
      const float mnew = fmaxf(mrow[r], mx);
      const float corr = __expf(mrow[r] - mnew);
      const float pe = valid ? __expf(sv - mnew) : 0.f;
      const float rs = half_sum16(pe);
      lrow[r] = lrow[r] * corr + rs;
      mrow[r] = mnew;
      p[r] = pe;
      #pragma unroll
      for (int nt = 0; nt < 8; ++nt) acc[nt][r] *= corr;
    }
    #pragma unroll
    for (int r = 0; r < 8; ++r) pl[r + hi * 8][ln] = p[r];
    __syncthreads();
    #pragma unroll
    for (int kk = 0; kk < 16; kk += 4) {
      v2f af = *(const v2f*)(&pl[ln][kk + hi * 2]);
      const float* vr0 = Vh + (size_t)(j0 + kk + hi * 2) * AHD_;
      const float* vr1 = vr0 + AHD_;
      #pragma unroll
      for (int nt = 0; nt < 8; ++nt) {
        v2f bf; bf.x = vr0[nt * 16 + ln]; bf.y = vr1[nt * 16 + ln];
        acc[nt] = wmma_f32(af, bf, acc[nt]);
      }
    }
    __syncthreads();
  }
  float* Ob = O + ((size_t)(b * S_) + i0) * H_ + h * AHD_;
  #pragma unroll
  for (int r = 0; r < 8; ++r) {
    const float inv = 1.0f / lrow[r];
    const int m = r + hi * 8;
    #pragma unroll
    for (int nt = 0; nt < 8; ++nt)
      Ob[(size_t)m * H_ + nt * 16 + ln] = acc[nt][r] * inv;
  }
}

// ---------------------------------------------------------------------------
// Elementwise / norm helpers
// ---------------------------------------------------------------------------
__global__ __launch_bounds__(256)
void rmsnorm2048(const float* __restrict__ in, const float* __restrict__ w,
                 float* __restrict__ out)
{
  __shared__ float red[256];
  const int t = blockIdx.x, tid = threadIdx.x;
  const float* row = in + (size_t)t * 2048;
  float s = 0.f;
  for (int c = tid; c < 2048; c += 256) { float v = row[c]; s += v * v; }
  red[tid] = s; __syncthreads();
  for (int st = 128; st > 0; st >>= 1) { if (tid < st) red[tid] += red[tid + st]; __syncthreads(); }
  const float r = rsqrtf(red[0] / 2048.f + EPS_);
  for (int c = tid; c < 2048; c += 256)
    out[(size_t)t * 2048 + c] = row[c] * r * w[c];
}

__global__ __launch_bounds__(128)
void rmsnorm512(float* ofw, const float* __restrict__ w)
{
  __shared__ float red[128];
  const int t = blockIdx.x, h = blockIdx.y, tid = threadIdx.x;
  float* base = ofw + (size_t)t * 2048 + h * 512;
  float s = 0.f;
  for (int c = tid; c < 512; c += 128) { float v = base[c]; s += v * v; }
  red[tid] = s; __syncthreads();
  for (int st = 64; st > 0; st >>= 1) { if (tid < st) red[tid] += red[tid + st]; __syncthreads(); }
  const float r = rsqrtf(red[0] / 512.f + EPS_);
  __syncthreads();
  for (int c = tid; c < 512; c += 128) base[c] = base[c] * r * w[c];
}

__global__ __launch_bounds__(128)
void l2norm512(float* fqkv)
{
  __shared__ float red[128];
  const int t = blockIdx.x, g = blockIdx.y, tid = threadIdx.x;
  const int part = g >> 2, h = g & 3;
  float* base = fqkv + (size_t)t * 6144 + part * 2048 + h * 512;
  float s = 0.f;
  for (int c = tid; c < 512; c += 128) { float v = base[c]; s += v * v; }
  red[tid] = s; __syncthreads();
  for (int st = 64; st > 0; st >>= 1) { if (tid < st) red[tid] += red[tid + st]; __syncthreads(); }
  const float inv = 1.f / fmaxf(sqrtf(red[0]), 1e-6f);
  __syncthreads();
  for (int c = tid; c < 512; c += 128) base[c] *= inv;
}

__global__ __launch_bounds__(256)
void lr_proj(const float* __restrict__ x, const float* __restrict__ lw,
             const float* __restrict__ lb, float* __restrict__ lrout)
{
  __shared__ float red[12][256];
  const int t = blockIdx.x, tid = threadIdx.x;
  const float* row = x + (size_t)t * 2048;
  float p[12];
  #pragma unroll
  for (int j = 0; j < 12; ++j) p[j] = 0.f;
  for (int c = tid; c < 2048; c += 256) {
    const float v = row[c];
    #pragma unroll
    for (int j = 0; j < 12; ++j) p[j] += v * lw[j * 2048 + c];
  }
  #pragma unroll
  for (int j = 0; j < 12; ++j) red[j][tid] = p[j];
  __syncthreads();
  for (int st = 128; st > 0; st >>= 1) {
    if (tid < st)
      #pragma unroll
      for (int j = 0; j < 12; ++j) red[j][tid] += red[j][tid + st];
    __syncthreads();
  }
  if (tid < 12) {
    const float z = red[tid][0] + lb[tid] + BASE_LR_INV_;
    const float sp = (z > 20.f) ? z : log1pf(__expf(z));
    lrout[(size_t)t * 12 + tid] = sp;
  }
}

__global__ __launch_bounds__(256)
void init_w(const float* __restrict__ w0i, const float* __restrict__ w1i,
            const float* __restrict__ w2i,
            float* __restrict__ W0, float* __restrict__ W1,
            float* __restrict__ W2)
{
  const size_t i = (size_t)blockIdx.x * 256 + threadIdx.x;
  if (i >= 8ull * 262144) return;
  const size_t h = (i / 262144) & 3;
  const size_t src = h * 262144 + (i % 262144);
  W0[i] = w0i[src]; W1[i] = w1i[src]; W2[i] = w2i[src];
}

__global__ __launch_bounds__(256)
void silu_mul(const float* a, const float* b, float* o, size_t n)
{
  const size_t i = (size_t)blockIdx.x * 256 + threadIdx.x;
  if (i < n) o[i] = silu_f(a[i]) * b[i];
}

__global__ __launch_bounds__(256)
void bprop(const float* __restrict__ dhid, const float* __restrict__ gk,
           const float* __restrict__ hbm,
           float* __restrict__ dgpre, float* __restrict__ dhbm, size_t n)
{
  const size_t i = (size_t)blockIdx.x * 256 + threadIdx.x;
  if (i < n) {
    const float g = gk[i];
    const float sg = 1.f / (1.f + __expf(-g));
    const float dh = dhid[i];
    dhbm[i] = dh * g * sg;                 // dhid * silu(gk)
    const float dgate = dh * hbm[i];
    dgpre[i] = dgate * sg * (1.f + g * (1.f - sg));
  }
}

// ---------------------------------------------------------------------------
extern "C" void kernel_launch(void* const* d_in, const int* in_sizes, int n_in,
                              void* d_out, int out_size, void* d_ws,
                              size_t ws_size, hipStream_t stream)
{
  (void)in_sizes; (void)n_in; (void)out_size;
  const float* hs    = (const float*)d_in[0];
  const float* aqkvw = (const float*)d_in[1];
  const float* aow   = (const float*)d_in[2];
  const float* qnw   = (const float*)d_in[3];
  const float* knw   = (const float*)d_in[4];
  const float* mnw   = (const float*)d_in[5];
  const float* mqkvw = (const float*)d_in[6];
  const float* mow   = (const float*)d_in[7];
  const float* lrw   = (const float*)d_in[8];
  const float* lrbp  = (const float*)d_in[9];
  const float* w0i   = (const float*)d_in[10];
  const float* w1i   = (const float*)d_in[11];
  const float* w2i   = (const float*)d_in[12];
  const float* tnw   = (const float*)d_in[13];
  float* out = (float*)d_out;
  float* ws  = (float*)d_ws;

  const size_t NEED = 81838080ull * 4ull; // 327 MB of f32 scratch
  if (ws_size < NEED) return;

  // workspace layout (floats) with lifetime-based reuse
  float* qkv  = ws;                       // [4096,6144]; reused as fqkv
  float* fqkv = qkv;
  float* qr   = ws + 25165824;            // [32,2048,128]; reused as o_fw
  float* ofw  = qr;
  float* kr   = ws + 33554432;            // rope k; reused: gq,t2,gk,hid
  float* gq   = kr;
  float* t2   = kr + 2097152;
  float* gk   = kr + 4194304;
  float* hid  = kr + 6291456;
  float* vr   = ws + 41943040;            // rope v; reused: dhid,dgpre,dhbm
  float* dhid = vr;
  float* dgpre= vr + 2097152;
  float* dhbm = vr + 4194304;
  float* ofl  = ws + 50331648;            // attention output, token-major
  float* aout = ws + 58720256;            // attn_out residual
  float* x    = ws + 67108864;            // rmsnorm(attn_out)
  float* lrb_ = ws + 75497472;            // [4096,12]
  float* W0   = ws + 75546624;            // [8,512,512] fast weights
  float* W1   = ws + 77643776;
  float* W2   = ws + 79740928;

  const dim3 gblk(32, 4);

  // 1. qkv = hs @ attn_qkv_w^T
  gemm_xwt<<<dim3(64, 96, 1), gblk, 0, stream>>>(
      hs, 2048, 0, 0, aqkvw, 2048, 0, 0, qkv, 6144, 0, 0, nullptr, 0, 2048, 0);
  // 2. rmsnorm(q,k) + rope + scatter to [b*h][s][d]
  rope_split<<<4096, 256, 0, stream>>>(qkv, qnw, knw, qr, kr, vr);
  // 3. sliding-window attention
  attn_swa<<<dim3(128, 16, 2), dim3(32), 0, stream>>>(qr, kr, vr, ofl);
  // 4. attn_out = o @ attn_o_w^T
  gemm_xwt<<<dim3(64, 32, 1), gblk, 0, stream>>>(
      ofl, 2048, 0, 0, aow, 2048, 0, 0, aout, 2048, 0, 0, nullptr, 0, 2048, 0);
  // 5. x = rmsnorm(attn_out) * mlp_norm_w
  rmsnorm2048<<<4096, 256, 0, stream>>>(aout, mnw, x);
  // 6. fqkv = silu(x @ mlp_qkv_w^T)   (overwrites qkv region; rope copies done)
  gemm_xwt<<<dim3(64, 96, 1), gblk, 0, stream>>>(
      x, 2048, 0, 0, mqkvw, 2048, 0, 0, fqkv, 6144, 0, 0, nullptr, 0, 2048, 1);
  // 7. lr = softplus(x @ lr_w^T + lr_b + inv_softplus(1e-3))
  lr_proj<<<4096, 256, 0, stream>>>(x, lrw, lrbp, lrb_);
  // 8. l2-normalize fq, fk per (token, head)
  l2norm512<<<dim3(4096, 8), 128, 0, stream>>>(fqkv);
  // 9. broadcast fast weights over batch
  init_w<<<8192, 256, 0, stream>>>(w0i, w1i, w2i, W0, W1, W2);

  // 10. TTT scan: 4 sequential chunks, batched over z = b*4+h
  const long Wsz = 262144;                       // 512*512 per (b,h)
  const long bsA = 2048L * 6144, hsA = 512;      // strided fqkv operands
  const long bsD = 4L * Wsz,    hsD = Wsz;       // dense per-bh scratch
  const dim3 g8(8, 8, 8);
  const size_t ne = 8ull * 262144;
  for (int ci = 0; ci < 4; ++ci) {
    const float* qc = fqkv + (size_t)ci * 512 * 6144;
    const float* kc = qc + 2048;
    const float* vc = qc + 4096;
    const float* l0 = lrb_ + (size_t)ci * 512 * 12;
    const float* l1 = l0 + 4;
    const float* l2 = l0 + 8;
    // apply: oc = (silu(q@W0^T) * (q@W2^T)) @ W1^T
    gemm_xwt<<<g8, gblk, 0, stream>>>(qc, 6144, bsA, hsA, W0, 512, bsD, hsD,
                                      gq, 512, bsD, hsD, nullptr, 0, 512, 0);
    gemm_xwt<<<g8, gblk, 0, stream>>>(qc, 6144, bsA, hsA, W2, 512, bsD, hsD,
                                      t2, 512, bsD, hsD, nullptr, 0, 512, 0);
    silu_mul<<<8192, 256, 0, stream>>>(gq, t2, gq, ne);
    gemm_xwt<<<g8, gblk, 0, stream>>>(gq, 512, bsD, hsD, W1, 512, bsD, hsD,
                                      ofw + (size_t)ci * 512 * 2048, 2048,
                                      2048L * 2048, 512, nullptr, 0, 512, 0);
    // update pre-activations: gk = k@W0^T, hbm = k@W2^T, hid = silu(gk)*hbm
    gemm_xwt<<<g8, gblk, 0, stream>>>(kc, 6144, bsA, hsA, W0, 512, bsD, hsD,
                                      gk, 512, bsD, hsD, nullptr, 0, 512, 0);
    gemm_xwt<<<g8, gblk, 0, stream>>>(kc, 6144, bsA, hsA, W2, 512, bsD, hsD,
                                      t2, 512, bsD, hsD, nullptr, 0, 512, 0);
    silu_mul<<<8192, 256, 0, stream>>>(gk, t2, hid, ne);
    // dhid = v @ W1
    gemm_xw<<<g8, gblk, 0, stream>>>(vc, 6144, bsA, hsA, W1, 512, bsD, hsD,
                                     dhid, 512, bsD, hsD, 512);
    // dhbm = dhid*silu(gk); dgpre = silu_backprop(dhid*hbm, gk)
    bprop<<<8192, 256, 0, stream>>>(dhid, gk, t2, dgpre, dhbm, ne);
    // rank-CHUNK fast-weight updates
    gemm_atb<<<g8, gblk, 0, stream>>>(vc, 6144, bsA, hsA, hid, 512, bsD, hsD,
                                      W1, 512, bsD, hsD, l1, nullptr,
                                      2048L * 12, 1, 12, 512);
    gemm_atb<<<g8, gblk, 0, stream>>>(dgpre, 512, bsD, hsD, kc, 6144, bsA, hsA,
                                      W0, 512, bsD, hsD, nullptr, l0,
                                      2048L * 12, 1, 12, 512);
    gemm_atb<<<g8, gblk, 0, stream>>>(dhbm, 512, bsD, hsD, kc, 6144, bsA, hsA,
                                      W2, 512, bsD, hsD, nullptr, l2,
                                      2048L * 12, 1, 12, 512);
  }

  // 11. ttt rmsnorm per (token, head)
  rmsnorm512<<<dim3(4096, 4), 128, 0, stream>>>(ofw, tnw);
  // 12. out = attn_out + o_fw @ mlp_o_w^T
  gemm_xwt<<<dim3(64, 32, 1), gblk, 0, stream>>>(
      ofw, 2048, 0, 0, mow, 2048, 0, 0, out, 2048, 0, 0, aout, 2048, 2048, 2);
}